// REVP_Transform_70428873720100
// MI455X (gfx1250) — compile-verified
//
#include <hip/hip_runtime.h>

#define H_T 680
#define W_T 680
#define NBINS (H_T * W_T)

typedef float v4f __attribute__((ext_vector_type(4)));

__device__ __forceinline__ void atom_addf(float* p, float v) {
    // agent-scope relaxed fp32 add -> global_atomic_add_f32 (no return)
    __hip_atomic_fetch_add(p, v, __ATOMIC_RELAXED, __HIP_MEMORY_SCOPE_AGENT);
}

__device__ __forceinline__ void accum_point(float u, float v,
                                            float f0, float f1, float f2, float f3,
                                            float wsc, float hsc,
                                            float* __restrict__ sums,
                                            float* __restrict__ counts) {
    // match jnp.round (round-half-even) then clip to [0, 679]
    int ui = (int)__builtin_rintf(u * wsc);
    int vi = (int)__builtin_rintf(v * hsc);
    ui = ui < 0 ? 0 : (ui > (W_T - 1) ? (W_T - 1) : ui);
    vi = vi < 0 ? 0 : (vi > (H_T - 1) ? (H_T - 1) : vi);
    int bin = vi * W_T + ui;
    atom_addf(&counts[bin], 1.0f);
    atom_addf(&sums[0 * NBINS + bin], f0);
    atom_addf(&sums[1 * NBINS + bin], f1);
    atom_addf(&sums[2 * NBINS + bin], f2);
    atom_addf(&sums[3 * NBINS + bin], f3);
}

// Zero the 4-channel sum image (d_out) and the counts buffer (ws).
__global__ void revp_zero_kernel(float* __restrict__ out, float* __restrict__ counts) {
    int i = blockIdx.x * blockDim.x + threadIdx.x;
    v4f z = {0.f, 0.f, 0.f, 0.f};
    if (i < NBINS)            // 4*NBINS floats = NBINS float4s
        ((v4f*)out)[i] = z;
    if (i < NBINS / 4)        // NBINS floats = NBINS/4 float4s
        ((v4f*)counts)[i] = z;
}

// Each thread consumes 4 points (96 bytes) via six 16B non-temporal loads.
__global__ void revp_scatter_kernel(const float* __restrict__ pts,
                                    const int* __restrict__ Hp,
                                    const int* __restrict__ Wp,
                                    float* __restrict__ sums,
                                    float* __restrict__ counts,
                                    int npts) {
    const float wsc = (float)W_T / (float)(*Wp);
    const float hsc = (float)H_T / (float)(*Hp);
    int t = blockIdx.x * blockDim.x + threadIdx.x;
    int base = t * 4;
    if (base >= npts) return;

    if (base + 3 < npts) {
        const v4f* p = (const v4f*)(pts + (size_t)base * 6);
        v4f a = __builtin_nontemporal_load(p + 0);
        v4f b = __builtin_nontemporal_load(p + 1);
        v4f c = __builtin_nontemporal_load(p + 2);
        v4f d = __builtin_nontemporal_load(p + 3);
        v4f e = __builtin_nontemporal_load(p + 4);
        v4f f = __builtin_nontemporal_load(p + 5);
        accum_point(a.x, a.y, a.z, a.w, b.x, b.y, wsc, hsc, sums, counts);
        accum_point(b.z, b.w, c.x, c.y, c.z, c.w, wsc, hsc, sums, counts);
        accum_point(d.x, d.y, d.z, d.w, e.x, e.y, wsc, hsc, sums, counts);
        accum_point(e.z, e.w, f.x, f.y, f.z, f.w, wsc, hsc, sums, counts);
    } else {
        // tail (not hit for N divisible by 4, but keep it correct)
        for (int k = base; k < npts; ++k) {
            const float* q = pts + (size_t)k * 6;
            accum_point(q[0], q[1], q[2], q[3], q[4], q[5], wsc, hsc, sums, counts);
        }
    }
}

// means = counts > 0 ? sums / counts : 0  (sums already 0 where count==0)
__global__ void revp_finalize_kernel(float* __restrict__ out,
                                     const float* __restrict__ counts) {
    int b = blockIdx.x * blockDim.x + threadIdx.x;
    if (b >= NBINS) return;
    float c = counts[b];
    if (c > 0.0f) {
        out[0 * NBINS + b] = out[0 * NBINS + b] / c;
        out[1 * NBINS + b] = out[1 * NBINS + b] / c;
        out[2 * NBINS + b] = out[2 * NBINS + b] / c;
        out[3 * NBINS + b] = out[3 * NBINS + b] / c;
    }
}

extern "C" void kernel_launch(void* const* d_in, const int* in_sizes, int n_in,
                              void* d_out, int out_size, void* d_ws, size_t ws_size,
                              hipStream_t stream) {
    const float* pts = (const float*)d_in[0];
    const int*   Hp  = (const int*)d_in[1];
    const int*   Wp  = (const int*)d_in[2];
    float* out    = (float*)d_out;           // 4 * NBINS sums -> means (channel-major image)
    float* counts = (float*)d_ws;            // NBINS floats of scratch

    const int npts = in_sizes[0] / 6;        // 8,000,000

    const int B = 256;
    int zero_blocks = (NBINS + B - 1) / B;
    revp_zero_kernel<<<zero_blocks, B, 0, stream>>>(out, counts);

    int nthreads = (npts + 3) / 4;
    int scat_blocks = (nthreads + B - 1) / B;
    revp_scatter_kernel<<<scat_blocks, B, 0, stream>>>(pts, Hp, Wp, out, counts, npts);

    int fin_blocks = (NBINS + B - 1) / B;
    revp_finalize_kernel<<<fin_blocks, B, 0, stream>>>(out, counts);
}